// InGramRelationLayer_64046552318120
// MI455X (gfx1250) — compile-verified
//
#include <hip/hip_runtime.h>
#include <math.h>

#define DIM_IN   128
#define DIM_OUT  128
#define NUM_HEAD 8
#define DIM_HID  16
#define NEG_SLOPE 0.2f
#define NCOLS 384   // [W1^T | W2^T | Waggr^T] columns

typedef __attribute__((ext_vector_type(2))) float v2f;
typedef __attribute__((ext_vector_type(8))) float v8f;

__device__ __forceinline__ float lrelu(float x) { return x >= 0.f ? x : NEG_SLOPE * x; }

// Float atomic max via monotone integer mapping (works for mixed signs,
// lowers to hardware global_atomic_max_i32 / global_atomic_min_u32).
__device__ __forceinline__ void atomicMaxF32(float* addr, float v) {
  if (v >= 0.f) atomicMax((int*)addr, __float_as_int(v));
  else          atomicMin((unsigned int*)addr, __float_as_uint(v));
}

// ---------------------------------------------------------------- init
__global__ void k_init(float* __restrict__ out, float* __restrict__ segmax,
                       float* __restrict__ segsum, int nOut, int nSeg) {
  int i = blockIdx.x * blockDim.x + threadIdx.x;
  if (i < nOut) out[i] = 0.f;
  if (i < nSeg) { segmax[i] = -3.402823466e38f; segsum[i] = 0.f; }
}

// ------------------------------------------- build B matrix (K=128 x N=384) + bias row
// B[k][n]: n<128   -> attn_proj_w[n][k]        (W1^T)
//          n<256   -> attn_proj_w[n-128][128+k] (W2^T)
//          else    -> aggr_proj_w[n-256][k]
__global__ void k_prep(const float* __restrict__ attn_w, const float* __restrict__ attn_b,
                       const float* __restrict__ aggr_w, const float* __restrict__ aggr_b,
                       float* __restrict__ Bmat, float* __restrict__ Bias) {
  int i = blockIdx.x * blockDim.x + threadIdx.x;
  if (i < DIM_IN * NCOLS) {
    int k = i / NCOLS, n = i % NCOLS;
    float v;
    if (n < 128)       v = attn_w[n * 256 + k];
    else if (n < 256)  v = attn_w[(n - 128) * 256 + 128 + k];
    else               v = aggr_w[(n - 256) * 128 + k];
    Bmat[i] = v;
  }
  if (i < NCOLS)
    Bias[i] = (i < 128) ? attn_b[i] : ((i < 256) ? 0.f : aggr_b[i - 256]);
}

// ------------------------------------------- fp32 WMMA GEMM: C[R x 384] = A[R x 128] * B
// One wave per block; each wave owns a 16(M) x 64(N) tile = 4 accumulators.
// A fragment (16x4 f32, ISA 7.12.2): lanes 0-15 -> M=lane, VGPR0/1 = K0/K1;
// lanes 16-31 -> M=lane-16, VGPR0/1 = K2/K3. B fragment mirrors (4x16, K split by half-wave).
__global__ void k_gemm(const float* __restrict__ A, const float* __restrict__ Bmat,
                       const float* __restrict__ Bias,
                       float* __restrict__ P1, float* __restrict__ P2,
                       float* __restrict__ Mout, int R) {
  const int lane = threadIdx.x & 31;
  const int half = lane >> 4;        // 0: K0/K1 side, 1: K2/K3 side
  const int l16  = lane & 15;
  const int m0 = blockIdx.x * 16;
  const int n0 = blockIdx.y * 64;

  int mrow = m0 + l16; if (mrow >= R) mrow = R - 1;   // clamp (R=20000 is exact anyway)
  const float* __restrict__ Arow = A + (size_t)mrow * DIM_IN;

  v8f acc0 = {}, acc1 = {}, acc2 = {}, acc3 = {};
  for (int k0 = 0; k0 < DIM_IN; k0 += 4) {
    const int ka = k0 + (half << 1);
    v2f a; a.x = Arow[ka]; a.y = Arow[ka + 1];
    const float* __restrict__ Bp = Bmat + (size_t)ka * NCOLS + n0 + l16;
    v2f b0; b0.x = Bp[0];  b0.y = Bp[NCOLS];
    v2f b1; b1.x = Bp[16]; b1.y = Bp[NCOLS + 16];
    v2f b2; b2.x = Bp[32]; b2.y = Bp[NCOLS + 32];
    v2f b3; b3.x = Bp[48]; b3.y = Bp[NCOLS + 48];
    acc0 = __builtin_amdgcn_wmma_f32_16x16x4_f32(false, a, false, b0, (short)0, acc0, false, false);
    acc1 = __builtin_amdgcn_wmma_f32_16x16x4_f32(false, a, false, b1, (short)0, acc1, false, false);
    acc2 = __builtin_amdgcn_wmma_f32_16x16x4_f32(false, a, false, b2, (short)0, acc2, false, false);
    acc3 = __builtin_amdgcn_wmma_f32_16x16x4_f32(false, a, false, b3, (short)0, acc3, false, false);
  }

  // C/D layout: VGPR i -> row m0 + i + 8*half, col = tile_base + l16
#pragma unroll
  for (int i = 0; i < 8; ++i) {
    const int row = m0 + i + (half << 3);
    if (row >= R) continue;
    float vals[4] = { acc0[i], acc1[i], acc2[i], acc3[i] };
#pragma unroll
    for (int j = 0; j < 4; ++j) {
      const int c = n0 + j * 16 + l16;          // block-uniform region (n0 multiple of 64)
      const float v = vals[j] + Bias[c];
      if (c < 128)      P1[(size_t)row * 128 + c]         = v;
      else if (c < 256) P2[(size_t)row * 128 + (c - 128)] = v;
      else              Mout[(size_t)row * 128 + (c - 256)] = v;
    }
  }
}

// ------------------------------------------- edge attention logits + segment max
// One wave per edge; lane handles dims [4*lane, 4*lane+4), head = lane>>2.
__global__ void k_attn(const int* __restrict__ trip, const float* __restrict__ P1,
                       const float* __restrict__ P2, const float* __restrict__ av,
                       const float* __restrict__ ab, float* __restrict__ attn,
                       float* __restrict__ segmax, int E) {
  const int lane = threadIdx.x & 31;
  const int e = blockIdx.x * (blockDim.x >> 5) + (threadIdx.x >> 5);
  if (e >= E) return;
  const int head = trip[3 * e], tail = trip[3 * e + 1], bin = trip[3 * e + 2];
  const int d0 = lane * 4;
  const float4 p1 = *(const float4*)(P1 + (size_t)head * 128 + d0);  // includes attn bias
  const float4 p2 = *(const float4*)(P2 + (size_t)tail * 128 + d0);
  const float4 w  = *(const float4*)(av + d0);
  float s = lrelu(p1.x + p2.x) * w.x + lrelu(p1.y + p2.y) * w.y +
            lrelu(p1.z + p2.z) * w.z + lrelu(p1.w + p2.w) * w.w;
  s += __shfl_xor(s, 1, 32);
  s += __shfl_xor(s, 2, 32);
  if ((lane & 3) == 0) {
    const int h = lane >> 2;
    const float r = s + lrelu(ab[bin * NUM_HEAD + h]);
    attn[(size_t)e * NUM_HEAD + h] = r;
    atomicMaxF32(segmax + head * NUM_HEAD + h, r);
  }
}

// ------------------------------------------- exp(x - segmax) + segment sum
__global__ void k_exp(const int* __restrict__ trip, float* __restrict__ attn,
                      const float* __restrict__ segmax, float* __restrict__ segsum, int EH) {
  int i = blockIdx.x * blockDim.x + threadIdx.x;
  if (i >= EH) return;
  const int e = i >> 3, h = i & 7;
  const int head = trip[3 * e];
  const float v = expf(attn[i] - segmax[head * NUM_HEAD + h]);
  attn[i] = v;
  atomicAdd(segsum + head * NUM_HEAD + h, v);
}

// ------------------------------------------- beta-weighted aggregation into out
__global__ void k_aggr(const int* __restrict__ trip, const float* __restrict__ attn,
                       const float* __restrict__ segsum, const float* __restrict__ Mmat,
                       float* __restrict__ out, int E) {
  const int lane = threadIdx.x & 31;
  const int e = blockIdx.x * (blockDim.x >> 5) + (threadIdx.x >> 5);
  if (e >= E) return;
  const int head = trip[3 * e], tail = trip[3 * e + 1];
  const int h = lane >> 2;
  const float beta = attn[(size_t)e * NUM_HEAD + h] /
                     (segsum[head * NUM_HEAD + h] + 1e-16f);
  const int d0 = lane * 4;
  const float4 m = *(const float4*)(Mmat + (size_t)tail * 128 + d0);
  float* o = out + (size_t)head * 128 + d0;
  atomicAdd(o + 0, beta * m.x);
  atomicAdd(o + 1, beta * m.y);
  atomicAdd(o + 2, beta * m.z);
  atomicAdd(o + 3, beta * m.w);
}

extern "C" void kernel_launch(void* const* d_in, const int* in_sizes, int n_in,
                              void* d_out, int out_size, void* d_ws, size_t ws_size,
                              hipStream_t stream) {
  const float* emb      = (const float*)d_in[0];
  const int*   trip     = (const int*)  d_in[1];
  const float* attn_w   = (const float*)d_in[2];
  const float* attn_b   = (const float*)d_in[3];
  const float* attn_bin = (const float*)d_in[4];
  const float* attn_vec = (const float*)d_in[5];
  const float* aggr_w   = (const float*)d_in[6];
  const float* aggr_b   = (const float*)d_in[7];

  const int R = in_sizes[0] / DIM_IN;   // 20000
  const int E = in_sizes[1] / 3;        // 640000

  // workspace layout (floats); totals ~52.7 MB
  float* ws     = (float*)d_ws;
  float* Bmat   = ws;                              // 128*384
  float* Bias   = Bmat + DIM_IN * NCOLS;           // 384
  float* P1     = Bias + NCOLS;                    // R*128  (16B-aligned: 198144 B offset)
  float* P2     = P1 + (size_t)R * 128;            // R*128
  float* Mm     = P2 + (size_t)R * 128;            // R*128
  float* attn   = Mm + (size_t)R * 128;            // E*8
  float* segmax = attn + (size_t)E * NUM_HEAD;     // R*8
  float* segsum = segmax + (size_t)R * NUM_HEAD;   // R*8
  float* out    = (float*)d_out;                   // R*128

  const int nOut = R * DIM_OUT, nSeg = R * NUM_HEAD;
  const int nInit = nOut > nSeg ? nOut : nSeg;
  k_init<<<(nInit + 255) / 256, 256, 0, stream>>>(out, segmax, segsum, nOut, nSeg);

  k_prep<<<(DIM_IN * NCOLS + 255) / 256, 256, 0, stream>>>(attn_w, attn_b, aggr_w, aggr_b, Bmat, Bias);

  dim3 gg((R + 15) / 16, NCOLS / 64);
  k_gemm<<<gg, 32, 0, stream>>>(emb, Bmat, Bias, P1, P2, Mm, R);

  k_attn<<<(E + 7) / 8, 256, 0, stream>>>(trip, P1, P2, attn_vec, attn_bin, attn, segmax, E);

  k_exp<<<(E * NUM_HEAD + 255) / 256, 256, 0, stream>>>(trip, attn, segmax, segsum, E * NUM_HEAD);

  k_aggr<<<(E + 7) / 8, 256, 0, stream>>>(trip, attn, segsum, Mm, out, E);
}